// YOLOLoss_32212254720459
// MI455X (gfx1250) — compile-verified
//
#include <hip/hip_runtime.h>
#include <hip/hip_bf16.h>

typedef float v2f __attribute__((ext_vector_type(2)));
typedef float v8f __attribute__((ext_vector_type(8)));

#define NUM_CLASSES 20
#define BATCH 16
#define NT 32

__device__ __forceinline__ float softplus_f(float x) {
    // bce(x, 0) = max(x,0) + log1p(exp(-|x|))
    return fmaxf(x, 0.0f) + log1pf(expf(-fabsf(x)));
}

// Full-wave (32-lane) sum using V_WMMA_F32_16X16X4_F32.
// A: lane partial in VGPR0 (K=0 for lanes 0-15, K=2 for lanes 16-31), VGPR1=0.
// B: all ones  =>  D[m,:] = p[m] + p[m+16].
// Per-lane fold of the 8 D VGPRs gives half-sums (rows 0-7 in lanes 0-15,
// rows 8-15 in lanes 16-31); one xor-16 shuffle completes the wave sum.
// Must be executed with EXEC all-ones (call from convergent code only).
__device__ __forceinline__ float wave_reduce_wmma(float v) {
    v2f a; a.x = v;    a.y = 0.0f;
    v2f b; b.x = 1.0f; b.y = 1.0f;
    v8f c = {};
    v8f d = __builtin_amdgcn_wmma_f32_16x16x4_f32(false, a, false, b,
                                                  (short)0, c, false, false);
    float s = (d[0] + d[1]) + (d[2] + d[3]) + (d[4] + d[5]) + (d[6] + d[7]);
    s += __shfl_xor(s, 16, 32);
    return s;
}

__global__ void yolo_init_ws(float* ws) {
    int i = threadIdx.x;
    if (i < 32) ws[i] = 0.0f;
}

// Dense streaming pass over the 15 "useful" channels per image (3 anchors x
// {4 box, 1 obj}).  Accumulates  S_box = sum pred^2  over box channels and
// S_objall = sum softplus(pred) over obj channels.
// Layout: p is (B, 75, H, W); channel c -> anchor a=c/25, k=c%25.
template <int H>
__global__ void __launch_bounds__(256)
yolo_dense_kernel(const float* __restrict__ p, float* __restrict__ acc) {
    constexpr int NPLANE = H * H;          // H*W elements per channel plane
    constexpr int NV     = NPLANE / 4;     // float4 vectors per plane
    constexpr int PLANES = BATCH * 15;     // useful planes total
    constexpr int NVEC   = PLANES * NV;

    float s_box = 0.0f;
    float s_obj = 0.0f;

    const int stride = gridDim.x * blockDim.x;
    for (int i = blockIdx.x * blockDim.x + threadIdx.x; i < NVEC; i += stride) {
        const int plane = i / NV;
        const int off4  = i - plane * NV;
        const int b     = plane / 15;
        const int r     = plane - b * 15;
        const int a     = r / 5;
        const int k     = r - a * 5;
        const size_t src = ((size_t)(b * 75 + a * 25 + k)) * NPLANE + (size_t)off4 * 4;
        const float4 v = *(const float4*)(p + src);
        if (k < 4) {
            s_box += v.x * v.x + v.y * v.y + v.z * v.z + v.w * v.w;
        } else {
            s_obj += softplus_f(v.x) + softplus_f(v.y) +
                     softplus_f(v.z) + softplus_f(v.w);
        }
    }

    // Convergent wave reduction (wmma), then LDS cross-wave, then atomics.
    __shared__ float red_box[8];
    __shared__ float red_obj[8];
    const int lane = threadIdx.x & 31;
    const int wave = threadIdx.x >> 5;
    const float wb = wave_reduce_wmma(s_box);
    const float wo = wave_reduce_wmma(s_obj);
    if (lane == 0) { red_box[wave] = wb; red_obj[wave] = wo; }
    __syncthreads();
    if (threadIdx.x == 0) {
        float tb = 0.0f, to = 0.0f;
        const int nw = blockDim.x >> 5;
        for (int w = 0; w < nw; ++w) { tb += red_box[w]; to += red_obj[w]; }
        atomicAdd(&acc[0], tb);
        atomicAdd(&acc[1], to);
    }
}

// Sparse correction pass: one wave per (scale, batch).  Computes per-target
// validity (IoU gate + triu shadowing dedup), then gathers the 25 channels of
// the matched anchor cell and accumulates correction terms.
__global__ void __launch_bounds__(32)
yolo_targets_kernel(const float* __restrict__ p0, const float* __restrict__ p1,
                    const float* __restrict__ p2, const float* __restrict__ tg,
                    float* __restrict__ ws) {
    const int s = blockIdx.x % 3;
    const int b = blockIdx.x / 3;
    const int t = threadIdx.x;

    const int   Htab[3] = {80, 40, 20};
    const float awt[3][3] = {{10.f/8.f,  16.f/8.f,  33.f/8.f},
                             {30.f/16.f, 62.f/16.f, 59.f/16.f},
                             {116.f/32.f,156.f/32.f,373.f/32.f}};
    const float aht[3][3] = {{13.f/8.f,  30.f/8.f,  23.f/8.f},
                             {61.f/16.f, 45.f/16.f,119.f/16.f},
                             {90.f/32.f, 198.f/32.f,326.f/32.f}};

    const int H = Htab[s];
    const int W = H;
    const float* p = (s == 0) ? p0 : (s == 1) ? p1 : p2;

    const float* tr = tg + ((size_t)b * NT + t) * 5;
    const int   cls = (int)tr[0];
    const float tx = tr[1], ty = tr[2], tw = tr[3], th = tr[4];

    int gx = (int)floorf(tx * (float)W);
    int gy = (int)floorf(ty * (float)H);
    gx = min(max(gx, 0), W - 1);
    gy = min(max(gy, 0), H - 1);

    // argmax IoU over 3 anchors (first max wins, matches jnp.argmax)
    float best = -1.0f; int ba = 0;
    #pragma unroll
    for (int a = 0; a < 3; ++a) {
        const float aw = awt[s][a], ah = aht[s][a];
        const float inter = fminf(tw, aw) * fminf(th, ah);
        const float iou = inter / (tw * th + aw * ah - inter + 1e-6f);
        if (iou > best) { best = iou; ba = a; }
    }
    bool valid = best > 0.3f;
    const int lin = (gy * W + gx) * 3 + ba;

    __shared__ int s_lin[NT];
    __shared__ int s_val[NT];
    s_lin[t] = lin;
    s_val[t] = valid ? 1 : 0;
    __syncthreads();
    bool shadowed = false;
    for (int u = t + 1; u < NT; ++u)
        if (s_lin[u] == lin && s_val[u]) shadowed = true;
    valid = valid && !shadowed;

    if (valid) {
        const float baw = awt[s][ba], bah = aht[s][ba];
        float tb[4];
        tb[0] = tx * (float)W - (float)gx;
        tb[1] = ty * (float)H - (float)gy;
        tb[2] = logf(tw / baw + 1e-6f);
        tb[3] = logf(th / bah + 1e-6f);

        const size_t npl = (size_t)H * W;
        const float* base = p + (size_t)(b * 75 + ba * 25) * npl
                              + (size_t)gy * W + gx;

        // box: (p-tb)^2 - p^2 = tb^2 - 2*p*tb
        float bc = 0.0f;
        #pragma unroll
        for (int k = 0; k < 4; ++k) {
            const float pv = base[(size_t)k * npl];
            bc += tb[k] * tb[k] - 2.0f * pv * tb[k];
        }
        // obj: bce(x,1) = softplus(x) - x ; also remove softplus from noobj sum
        const float po  = base[4 * npl];
        const float spo = softplus_f(po);
        // cls: sum_j softplus(p_j) - p_cls
        float cs = 0.0f;
        #pragma unroll
        for (int j = 0; j < NUM_CLASSES; ++j) {
            const float pc = base[(size_t)(5 + j) * npl];
            cs += softplus_f(pc) - ((j == cls) ? pc : 0.0f);
        }

        float* acc = ws + s * 8;
        atomicAdd(&acc[2], 1.0f);       // n_obj
        atomicAdd(&acc[3], bc);         // box correction
        atomicAdd(&acc[4], spo - po);   // obj bce sum
        atomicAdd(&acc[5], spo);        // softplus at obj cells (noobj fixup)
        atomicAdd(&acc[6], cs);         // cls bce sum
    }
}

__global__ void yolo_final_kernel(const float* __restrict__ ws,
                                  float* __restrict__ out) {
    if (threadIdx.x == 0 && blockIdx.x == 0) {
        const int Htab[3] = {80, 40, 20};
        float total = 0.0f;
        #pragma unroll
        for (int s = 0; s < 3; ++s) {
            const float* a = ws + s * 8;
            const float cells   = (float)BATCH * (float)(Htab[s] * Htab[s]) * 3.0f;
            const float n_obj   = a[2] + 1e-6f;
            const float n_noobj = (cells - a[2]) + 1e-6f;
            const float box   = (a[0] + a[3]) / n_obj;
            const float obj   = a[4] / n_obj;
            const float noobj = (a[1] - a[5]) / n_noobj;
            const float cl    = a[6] / n_obj;
            total += 0.05f * box + 1.5f * (obj + 0.5f * noobj) + 0.15f * cl;
        }
        out[0] = total;
    }
}

extern "C" void kernel_launch(void* const* d_in, const int* in_sizes, int n_in,
                              void* d_out, int out_size, void* d_ws, size_t ws_size,
                              hipStream_t stream) {
    const float* p0 = (const float*)d_in[0];
    const float* p1 = (const float*)d_in[1];
    const float* p2 = (const float*)d_in[2];
    const float* tg = (const float*)d_in[3];
    float* out = (float*)d_out;
    float* ws  = (float*)d_ws;

    yolo_init_ws<<<1, 32, 0, stream>>>(ws);

    // vectors per scale: BATCH*15*(H*H/4); ~4 float4 per thread
    const int nv0 = BATCH * 15 * (80 * 80 / 4);
    const int nv1 = BATCH * 15 * (40 * 40 / 4);
    const int nv2 = BATCH * 15 * (20 * 20 / 4);
    const int g0 = (nv0 + 1023) / 1024;
    const int g1 = (nv1 + 1023) / 1024;
    const int g2 = (nv2 + 1023) / 1024;

    yolo_dense_kernel<80><<<g0, 256, 0, stream>>>(p0, ws + 0);
    yolo_dense_kernel<40><<<g1, 256, 0, stream>>>(p1, ws + 8);
    yolo_dense_kernel<20><<<g2, 256, 0, stream>>>(p2, ws + 16);

    yolo_targets_kernel<<<48, 32, 0, stream>>>(p0, p1, p2, tg, ws);

    yolo_final_kernel<<<1, 1, 0, stream>>>(ws, out);
}